// Concat_input_stacom_45964740001829
// MI455X (gfx1250) — compile-verified
//
#include <hip/hip_runtime.h>
#include <hip/hip_bf16.h>

// ---------------- problem constants (from the reference) ----------------
#define NPF   150000                 // voxels per frame
#define CCH   128                    // channels
#define NFR   4                      // frames
#define NT    (NPF * NFR)            // 600000 total rows
#define BB    4
#define DD    13
#define HH    128
#define WW    128
#define KEYS  (BB * DD * HH * WW)    // 851968 dense voxel cells
#define SCAN_TILE   2048             // flags per scan block (256 thr * 8)
#define SCAN_BLOCKS (KEYS / SCAN_TILE) // 416 exactly

static __device__ __forceinline__ unsigned lds_byte_addr(const void* p) {
    // generic pointer into LDS aperture: low 32 bits == wave-relative LDS offset
    return (unsigned)(unsigned long long)p;
}

static __device__ __forceinline__ void async_load_32B_to_lds(unsigned lds, const unsigned* gp) {
    // CDNA5 async DMA: global -> LDS, 2 x b128 per lane, tracked by ASYNCcnt
    asm volatile(
        "global_load_async_to_lds_b128 %0, %1, off\n\t"
        "global_load_async_to_lds_b128 %0, %1, off offset:16"
        :: "v"(lds), "v"(gp) : "memory");
}

static __device__ __forceinline__ void wait_asynccnt0() {
    asm volatile("s_wait_asynccnt 0x0" ::: "memory");
}

// ---------------- 1) flag presence of each key ----------------
__global__ void flag_kernel(const int4* __restrict__ i0, const int4* __restrict__ i1,
                            const int4* __restrict__ i2, const int4* __restrict__ i3,
                            unsigned* __restrict__ flags) {
    int i = blockIdx.x * blockDim.x + threadIdx.x;
    if (i >= NT) return;
    int f = i / NPF;
    int r = i - f * NPF;
    const int4* ip = (f == 0) ? i0 : (f == 1) ? i1 : (f == 2) ? i2 : i3;
    int4 v = ip[r];
    unsigned key = ((unsigned)((v.x * DD + v.y) * HH + v.z)) * WW + (unsigned)v.w;
    flags[key] = 1u;                 // duplicate writes of identical value: benign
}

// ---------------- 2a) per-block sums of flags (async LDS staging) ----------------
__global__ void scan_partials_kernel(const unsigned* __restrict__ flags,
                                     unsigned* __restrict__ partials) {
    __shared__ unsigned tile[SCAN_TILE];
    __shared__ unsigned red[256];
    int t = threadIdx.x;
    const unsigned* gp = flags + (size_t)blockIdx.x * SCAN_TILE + t * 8;
    async_load_32B_to_lds(lds_byte_addr(tile + t * 8), gp);
    wait_asynccnt0();
    unsigned s = 0;
#pragma unroll
    for (int k = 0; k < 8; ++k) s += tile[t * 8 + k];
    red[t] = s;
    __syncthreads();
    for (int off = 128; off > 0; off >>= 1) {
        if (t < off) red[t] += red[t + off];
        __syncthreads();
    }
    if (t == 0) partials[blockIdx.x] = red[0];
}

// ---------------- 2b) single-block scan of 416 partials, emit n_unique ----------------
__global__ void scan_spine_kernel(unsigned* __restrict__ partials,
                                  float* __restrict__ nuniq_out) {
    __shared__ unsigned sc[512];
    int t = threadIdx.x;
    unsigned v = (t < SCAN_BLOCKS) ? partials[t] : 0u;
    sc[t] = v;
    __syncthreads();
    for (int off = 1; off < 512; off <<= 1) {
        unsigned y = (t >= off) ? sc[t - off] : 0u;
        unsigned x = sc[t];
        __syncthreads();
        sc[t] = x + y;
        __syncthreads();
    }
    if (t < SCAN_BLOCKS) partials[t] = sc[t] - v;      // exclusive
    if (t == SCAN_BLOCKS - 1) nuniq_out[0] = (float)sc[t];  // total unique count
}

// ---------------- 2c) write rank[key] = exclusive prefix of flags ----------------
__global__ void scan_rank_kernel(unsigned* __restrict__ flags,
                                 const unsigned* __restrict__ partials) {
    __shared__ unsigned tile[SCAN_TILE];
    __shared__ unsigned sc[256];
    int t = threadIdx.x;
    unsigned* gp = flags + (size_t)blockIdx.x * SCAN_TILE + t * 8;
    async_load_32B_to_lds(lds_byte_addr(tile + t * 8), gp);
    wait_asynccnt0();
    unsigned vals[8];
    unsigned s = 0;
#pragma unroll
    for (int k = 0; k < 8; ++k) { vals[k] = tile[t * 8 + k]; s += vals[k]; }
    sc[t] = s;
    __syncthreads();
    for (int off = 1; off < 256; off <<= 1) {
        unsigned y = (t >= off) ? sc[t - off] : 0u;
        unsigned x = sc[t];
        __syncthreads();
        sc[t] = x + y;
        __syncthreads();
    }
    unsigned run = partials[blockIdx.x] + sc[t] - s;   // block offset + exclusive
    unsigned out[8];
#pragma unroll
    for (int k = 0; k < 8; ++k) { out[k] = run; run += vals[k]; }
    uint4* o = (uint4*)gp;                             // overwrite flags with ranks
    o[0] = make_uint4(out[0], out[1], out[2], out[3]);
    o[1] = make_uint4(out[4], out[5], out[6], out[7]);
}

// ---------------- 3) scatter-add features, write merged indices ----------------
__global__ void scatter_kernel(const float* __restrict__ f0, const float* __restrict__ f1,
                               const float* __restrict__ f2, const float* __restrict__ f3,
                               const int4* __restrict__ i0, const int4* __restrict__ i1,
                               const int4* __restrict__ i2, const int4* __restrict__ i3,
                               const unsigned* __restrict__ rank,
                               float* __restrict__ feat_out, float* __restrict__ idx_out) {
    int wave = (int)((blockIdx.x * blockDim.x + threadIdx.x) >> 5);  // one wave32 per row
    int lane = threadIdx.x & 31;
    if (wave >= NT) return;
    int f = wave / NPF;
    int r = wave - f * NPF;
    const int4*  ip = (f == 0) ? i0 : (f == 1) ? i1 : (f == 2) ? i2 : i3;
    const float* fp = (f == 0) ? f0 : (f == 1) ? f1 : (f == 2) ? f2 : f3;

    const float* row = fp + (size_t)r * CCH;
    __builtin_prefetch(row, 0, 0);                     // global_prefetch_b8

    int4 v = ip[r];
    unsigned key = ((unsigned)((v.x * DD + v.y) * HH + v.z)) * WW + (unsigned)v.w;
    unsigned u = rank[key];                            // unique id in sorted-key order

    float4 d = ((const float4*)row)[lane];             // 32 lanes * float4 == 128 ch
    float* dst = feat_out + (size_t)u * CCH + lane * 4;
    atomicAdd(dst + 0, d.x);                           // global_atomic_add_f32 @ L2
    atomicAdd(dst + 1, d.y);
    atomicAdd(dst + 2, d.z);
    atomicAdd(dst + 3, d.w);

    if (lane == 0) {                                   // idempotent across duplicates
        float* io = idx_out + (size_t)u * 4;
        io[0] = (float)v.x; io[1] = (float)v.y; io[2] = (float)v.z; io[3] = (float)v.w;
    }
}

// ---------------- host side ----------------
extern "C" void kernel_launch(void* const* d_in, const int* in_sizes, int n_in,
                              void* d_out, int out_size, void* d_ws, size_t ws_size,
                              hipStream_t stream) {
    // setup_inputs() insertion order: feat0, idx0, feat1, idx1, feat2, idx2, feat3, idx3
    const float* f0 = (const float*)d_in[0];
    const int4*  i0 = (const int4*)d_in[1];
    const float* f1 = (const float*)d_in[2];
    const int4*  i1 = (const int4*)d_in[3];
    const float* f2 = (const float*)d_in[4];
    const int4*  i2 = (const int4*)d_in[5];
    const float* f3 = (const float*)d_in[6];
    const int4*  i3 = (const int4*)d_in[7];

    const size_t ws_needed = (size_t)(KEYS + SCAN_BLOCKS + 64) * sizeof(unsigned);
    if (ws_size < ws_needed) return;

    unsigned* flags    = (unsigned*)d_ws;          // KEYS entries: flags then ranks
    unsigned* partials = flags + KEYS;             // SCAN_BLOCKS entries

    float* feat_out = (float*)d_out;               // [NT, 128]
    float* idx_out  = feat_out + (size_t)NT * CCH; // [NT, 4]
    float* n_out    = idx_out + (size_t)NT * 4;    // scalar n_unique

    // zero outputs (atomics accumulate) and the flag grid, every call
    hipMemsetAsync(d_out, 0, (size_t)out_size * sizeof(float), stream);
    hipMemsetAsync(flags, 0, (size_t)KEYS * sizeof(unsigned), stream);

    flag_kernel<<<(NT + 255) / 256, 256, 0, stream>>>(i0, i1, i2, i3, flags);
    scan_partials_kernel<<<SCAN_BLOCKS, 256, 0, stream>>>(flags, partials);
    scan_spine_kernel<<<1, 512, 0, stream>>>(partials, n_out);
    scan_rank_kernel<<<SCAN_BLOCKS, 256, 0, stream>>>(flags, partials);
    scatter_kernel<<<NT / 8, 256, 0, stream>>>(f0, f1, f2, f3, i0, i1, i2, i3,
                                               flags, feat_out, idx_out);
}